// GGXSampler_65309272703284
// MI455X (gfx1250) — compile-verified
//
#include <hip/hip_runtime.h>
#include <hip/hip_bf16.h>
#include <cstdint>
#include <math.h>

#define PI_F 3.14159265358979323846f

struct F3 { float x, y, z; };

__device__ __forceinline__ F3 cross3(F3 a, F3 b) {
  return F3{ a.y*b.z - a.z*b.y, a.z*b.x - a.x*b.z, a.x*b.y - a.y*b.x };
}
__device__ __forceinline__ float dot3(F3 a, F3 b) { return a.x*b.x + a.y*b.y + a.z*b.z; }
// matches reference: x / (||x|| + 1e-8)
__device__ __forceinline__ F3 nrm3(F3 v) {
  float inv = 1.0f / (sqrtf(dot3(v, v)) + 1e-8f);
  return F3{ v.x*inv, v.y*inv, v.z*inv };
}

// 8 wave32 per block, one shading point per wave, 4 *contiguous* samples per
// lane (SS = 128 = 32 lanes * 4), so every global access is a b128 op.
// SS is a compile-time sample count (0 = runtime fallback via Sdyn).
template <int SS>
__global__ __launch_bounds__(256) void ggx_sample_kernel(
    const float* __restrict__ viewdir,
    const float* __restrict__ normal,
    const float* __restrict__ roughness,
    const float* __restrict__ angs,
    const int*   __restrict__ ray_mask,
    const int*   __restrict__ num_samples_p,
    float* __restrict__ outL,
    float* __restrict__ outMip,
    int Bn, int Sdyn)
{
  const int S = (SS > 0) ? SS : Sdyn;

  // 8 waves * 128 samples * 2 floats = 8 KB
  __shared__ __attribute__((aligned(16))) float s_angs[8][256];

  const int tid  = threadIdx.x;
  const int w    = tid >> 5;      // wave index within block
  const int lane = tid & 31;
  const int b    = blockIdx.x * 8 + w;
  if (b >= Bn) return;

  // ---------------------------------------------------------------
  // CDNA5 async DMA: stream this point's angs row (S*8 bytes) from
  // HBM straight into LDS; overlaps with the frame setup below.
  // Base address readfirstlane'd once into an SGPR pair; no exec
  // masking needed (S multiple of 64 -> every lane has a chunk).
  // For SS=128 this is exactly two straight-line issues per lane.
  // ---------------------------------------------------------------
  {
    uint64_t g  = (uint64_t)(uintptr_t)angs + (uint64_t)b * (uint64_t)S * 8u;
    uint32_t lo = (uint32_t)__builtin_amdgcn_readfirstlane((int)(g & 0xffffffffu));
    uint32_t hi = (uint32_t)__builtin_amdgcn_readfirstlane((int)(g >> 32));
    uint64_t saddr = ((uint64_t)hi << 32) | (uint64_t)lo;
    unsigned lbase = (unsigned)(uintptr_t)(&s_angs[w][0]); // low 32b = LDS offset
    const int chunks = (S * 8) >> 9;        // 512B per wave pass (16B/lane)
    unsigned c = (unsigned)(lane * 16);
#pragma unroll
    for (int i = 0; i < ((SS > 0) ? ((SS * 8) >> 9) : 1); ++i) {
      unsigned laddr = lbase + c;
      asm volatile("global_load_async_to_lds_b128 %0, %1, %2"
                   :: "v"(laddr), "v"(c), "s"(saddr)
                   : "memory");
      c += 512u;
    }
    if (SS == 0) {   // runtime fallback: remaining chunks
      for (int i = 1; i < chunks; ++i) {
        unsigned laddr = lbase + c;
        asm volatile("global_load_async_to_lds_b128 %0, %1, %2"
                     :: "v"(laddr), "v"(c), "s"(saddr)
                     : "memory");
        c += 512u;
      }
    }
  }

  // ------------------- per-point frame setup ---------------------
  F3 n = { normal[b*3+0],  normal[b*3+1],  normal[b*3+2] };
  F3 v = { viewdir[b*3+0], viewdir[b*3+1], viewdir[b*3+2] };
  float r = roughness[b];

  F3 up      = (n.z < 0.9f) ? F3{0.f,0.f,1.f} : F3{1.f,0.f,0.f};
  F3 tangent = nrm3(cross3(up, n));
  F3 bitang  = nrm3(cross3(n, tangent));
  // V_l = rwb @ v ; rwb rows = tangent, bitangent, normal
  F3 Vl = { dot3(tangent, v), dot3(bitang, v), dot3(n, v) };
  F3 Vs = nrm3(F3{ r*Vl.x, r*Vl.y, Vl.z });
  // cross(Vs, [0,0,1]) = (Vs.y, -Vs.x, 0)
  F3 T1 = (Vs.z < 0.999f) ? nrm3(F3{ Vs.y, -Vs.x, 0.0f }) : F3{1.f,0.f,0.f};
  F3 T2 = nrm3(cross3(T1, Vs));
  const float z = Vs.z;
  const float a = fminf(1.0f / fmaxf(1.0f + z, 1e-5f), 1e4f);

  const float NdotV     = fminf(fmaxf(fabsf(dot3(n, v)), 1e-8f), 1.0f);
  const float rc        = fmaxf(r, 0.001f);
  const float a2        = rc * rc;
  const float a2m1      = a2 - 1.0f;
  const float log_NdotV = __logf(NdotV);
  const float log_r     = __logf(fmaxf(r, 1e-5f));
  const float log_num   = __logf((float)(*num_samples_p));
  const float inv_a     = 1.0f / a;
  const float inv_1ma   = 1.0f / (1.0f - a);   // dead lane of the select if a==1

  // DMA must have landed before the LDS tile is read.
  asm volatile("s_wait_asynccnt 0x0" ::: "memory");

  // ----------------------- sample loop ---------------------------
  const float4* lrow = reinterpret_cast<const float4*>(&s_angs[w][0]);
  for (int g0 = 0; g0 < S; g0 += 128) {   // single iteration when SS==128
    const int s0 = g0 + lane * 4;                       // 4 contiguous samples
    const size_t idx0 = (size_t)b * (size_t)S + (size_t)s0;

    // 32B of angs per lane: two ds_load_b128
    const float4 aA = lrow[(g0 >> 1) + lane * 2 + 0];   // u1_0,u2_0,u1_1,u2_1
    const float4 aB = lrow[(g0 >> 1) + lane * 2 + 1];   // u1_2,u2_2,u1_3,u2_3
    // 4 masks in one global_load_b128
    const int4 mv = *reinterpret_cast<const int4*>(ray_mask + idx0);

    const float u1s[4] = { aA.x, aA.z, aB.x, aB.z };
    const float u2s[4] = { aA.y, aA.w, aB.y, aB.w };
    const int   ms[4]  = { mv.x, mv.y, mv.z, mv.w };

    F3    Lv[4];
    float mips[4];
#pragma unroll
    for (int j = 0; j < 4; ++j) {
      const float u1 = u1s[j];
      const float u2 = u2s[j];

      const float rs  = sqrtf(u1);
      const bool  lt  = u2 < a;
      const float phi = lt ? (u2 * inv_a) * PI_F
                           : fmaf((u2 - a) * inv_1ma, PI_F, PI_F);
      const float P1 = rs * __cosf(phi);
      const float P2 = rs * __sinf(phi) * (lt ? 1.0f : z);
      const float P3 = sqrtf(fmaxf(1.0f - P1*P1 - P2*P2, 0.0f));

      F3 Ns = { P1*T1.x + P2*T2.x + P3*Vs.x,
                P1*T1.y + P2*T2.y + P3*Vs.y,
                P1*T1.z + P2*T2.z + P3*Vs.z };
      F3 Hl = nrm3(F3{ r*Ns.x, r*Ns.y, fmaxf(Ns.z, 0.0f) });
      // H = rwb^T @ H_l
      F3 H  = { Hl.x*tangent.x + Hl.y*bitang.x + Hl.z*n.x,
                Hl.x*tangent.y + Hl.y*bitang.y + Hl.z*n.y,
                Hl.x*tangent.z + Hl.y*bitang.z + Hl.z*n.z };

      const float VdotH = dot3(v, H);
      F3 L = { 2.0f*VdotH*H.x - v.x,
               2.0f*VdotH*H.y - v.y,
               2.0f*VdotH*H.z - v.z };

      const float NdotH = fminf(fmaxf(dot3(H, n) + 0.001f, 1e-8f), 1.0f);
      const float HdotV = fabsf(VdotH);
      const float Dv    = a2 / fmaf(NdotH*NdotH, a2m1, 1.0f);
      const float D     = fminf(Dv*Dv, 1.0f);
      const float lpdf  = __logf(fmaxf(D, 1e-5f)) + __logf(fmaxf(HdotV, 1e-5f))
                          - log_NdotV - log_r;
      float mip = -log_num - lpdf;

      if (!ms[j]) { L = F3{0.f,0.f,0.f}; mip = 0.0f; }
      Lv[j]   = L;
      mips[j] = mip;
    }

    // 48B of L per lane -> 3 x global_store_b128, fully contiguous per wave
    float4* po = reinterpret_cast<float4*>(outL + idx0 * 3u);
    po[0] = make_float4(Lv[0].x, Lv[0].y, Lv[0].z, Lv[1].x);
    po[1] = make_float4(Lv[1].y, Lv[1].z, Lv[2].x, Lv[2].y);
    po[2] = make_float4(Lv[2].z, Lv[3].x, Lv[3].y, Lv[3].z);
    // 16B of mipval per lane -> 1 x global_store_b128
    *reinterpret_cast<float4*>(outMip + idx0) =
        make_float4(mips[0], mips[1], mips[2], mips[3]);
  }
}

extern "C" void kernel_launch(void* const* d_in, const int* in_sizes, int n_in,
                              void* d_out, int out_size, void* d_ws, size_t ws_size,
                              hipStream_t stream) {
  // setup_inputs() order:
  // 0: refdirs (B,3)  -- unused by the reference computation
  // 1: viewdir (B,3)
  // 2: normal  (B,3)
  // 3: roughness (B,)
  // 4: angs (B,S,2)
  // 5: ray_mask (B,S)  int
  // 6: num_samples     int scalar
  const float* viewdir   = (const float*)d_in[1];
  const float* normal    = (const float*)d_in[2];
  const float* roughness = (const float*)d_in[3];
  const float* angs      = (const float*)d_in[4];
  const int*   ray_mask  = (const int*)d_in[5];
  const int*   num_samp  = (const int*)d_in[6];

  const int Bn = in_sizes[3];
  const int S  = in_sizes[5] / Bn;

  float* outL   = (float*)d_out;                       // (B,S,3) flat
  float* outMip = outL + (size_t)Bn * (size_t)S * 3u;  // (B,S) flat

  const int blocks = (Bn + 7) / 8;                     // 8 points per block
  if (S == 128) {
    ggx_sample_kernel<128><<<blocks, 256, 0, stream>>>(
        viewdir, normal, roughness, angs, ray_mask, num_samp,
        outL, outMip, Bn, S);
  } else {
    ggx_sample_kernel<0><<<blocks, 256, 0, stream>>>(
        viewdir, normal, roughness, angs, ray_mask, num_samp,
        outL, outMip, Bn, S);
  }
}